// talinear_53867479826614
// MI455X (gfx1250) — compile-verified
//
#include <hip/hip_runtime.h>
#include <hip/hip_bf16.h>

typedef __attribute__((ext_vector_type(2))) float v2f;
typedef __attribute__((ext_vector_type(8))) float v8f;

#define B_DIM 64
#define G_DIM 307
#define E_DIM 64
#define T_DIM 12
#define K_DIM 768   // E*T
#define N_DIM 768   // E*T
#define WBANK_ELEMS (768 * 768)  // T*E*E*T

#define BM 128      // rows (g) per block
#define BN 64       // cols (n) per block
#define KC 64       // k-chunk staged in LDS
#define LDA 68      // padded LDS stride (floats) for A
#define LDB 68      // padded LDS stride (floats) for Bt

#define LEAKY_ALPHA 0.3f

__global__ __launch_bounds__(256)
void talinear_wmma_kernel(const float* __restrict__ inputs,
                          const int*   __restrict__ week,
                          const int*   __restrict__ date,
                          const float* __restrict__ week_w,
                          const float* __restrict__ week_b,
                          const float* __restrict__ day_w,
                          const float* __restrict__ day_b,
                          float* __restrict__ out) {
    __shared__ float As[BM * LDA];      // A[row][k], row-major, padded
    __shared__ float Bt[BN * LDB];      // B transposed: Bt[n][k], padded

    const int n0 = blockIdx.x * BN;
    const int g0 = blockIdx.y * BM;
    const int b  = blockIdx.z;

    const int tid  = threadIdx.x;
    const int lane = tid & 31;
    const int wave = tid >> 5;
    const int wm   = wave & 3;          // 4 row groups of 32
    const int wn   = wave >> 2;         // 2 col groups of 32

    const int wk = week[b];
    const int dt = date[b];

    const float* Wbank0 = week_w + (size_t)wk * WBANK_ELEMS;
    const float* Wbank1 = day_w  + (size_t)dt * WBANK_ELEMS;
    const float* Ab     = inputs + (size_t)b * G_DIM * K_DIM;

    v8f c00 = {}, c01 = {}, c10 = {}, c11 = {};

    const int mrow = lane & 15;                 // M (or N) index within 16
    const int koff = (lane >> 4) << 1;          // K offset: 0 (lanes 0-15) / 2 (lanes 16-31)

    for (int bank = 0; bank < 2; ++bank) {
        const float* W = (bank == 0) ? Wbank0 : Wbank1;
        for (int kc = 0; kc < K_DIM; kc += KC) {
            __syncthreads();
            // ---- stage A: BM x KC floats (2048 float4, 8 per thread) ----
            #pragma unroll
            for (int i = 0; i < 8; ++i) {
                int idx = tid + i * 256;            // 0..2047
                int row = idx >> 4;                 // KC/4 = 16 float4 per row
                int c4  = (idx & 15) << 2;          // float column
                int g   = g0 + row;
                if (g >= G_DIM) g = G_DIM - 1;      // clamp (rows never stored)
                const float4 v = *(const float4*)(Ab + (size_t)g * K_DIM + kc + c4);
                *(float4*)&As[row * LDA + c4] = v;  // LDA*row is multiple of 4
            }
            // ---- stage B transposed: KC x BN floats (1024 float4, 4 per thread) ----
            // A's k = e*T + t  maps to weight row (t*E + e); n is contiguous.
            #pragma unroll
            for (int i = 0; i < 4; ++i) {
                int idx = tid + i * 256;            // 0..1023
                int kr  = idx >> 4;                 // k within chunk, 0..63
                int c4  = (idx & 15) << 2;          // n column
                int k   = kc + kr;
                int e   = k / T_DIM;
                int t   = k - e * T_DIM;
                const float4 v =
                    *(const float4*)(W + (size_t)(t * E_DIM + e) * N_DIM + n0 + c4);
                Bt[(c4 + 0) * LDB + kr] = v.x;
                Bt[(c4 + 1) * LDB + kr] = v.y;
                Bt[(c4 + 2) * LDB + kr] = v.z;
                Bt[(c4 + 3) * LDB + kr] = v.w;
            }
            __syncthreads();
            // ---- compute: 16 k-groups of 4, 4 WMMAs each ----
            #pragma unroll
            for (int kg = 0; kg < KC; kg += 4) {
                v2f a0 = *(const v2f*)&As[(wm * 32 +      mrow) * LDA + kg + koff];
                v2f a1 = *(const v2f*)&As[(wm * 32 + 16 + mrow) * LDA + kg + koff];
                v2f b0 = *(const v2f*)&Bt[(wn * 32 +      mrow) * LDB + kg + koff];
                v2f b1 = *(const v2f*)&Bt[(wn * 32 + 16 + mrow) * LDB + kg + koff];
                c00 = __builtin_amdgcn_wmma_f32_16x16x4_f32(false, a0, false, b0,
                                                            (short)0, c00, false, false);
                c01 = __builtin_amdgcn_wmma_f32_16x16x4_f32(false, a0, false, b1,
                                                            (short)0, c01, false, false);
                c10 = __builtin_amdgcn_wmma_f32_16x16x4_f32(false, a1, false, b0,
                                                            (short)0, c10, false, false);
                c11 = __builtin_amdgcn_wmma_f32_16x16x4_f32(false, a1, false, b1,
                                                            (short)0, c11, false, false);
            }
        }
    }

    // ---- epilogue: (acc + wb + db) * 0.5, LeakyReLU ----
    const float* wb = week_b + (size_t)wk * G_DIM * N_DIM;
    const float* db = day_b  + (size_t)dt * G_DIM * N_DIM;
    const int mbase = (lane >> 4) << 3;         // C layout: lanes 16-31 -> M = v + 8
    const int ncol  = lane & 15;

    #pragma unroll
    for (int si = 0; si < 2; ++si) {
        #pragma unroll
        for (int sj = 0; sj < 2; ++sj) {
            const v8f acc = (si == 0) ? ((sj == 0) ? c00 : c01)
                                      : ((sj == 0) ? c10 : c11);
            const int ng = n0 + wn * 32 + sj * 16 + ncol;
            #pragma unroll
            for (int v = 0; v < 8; ++v) {
                const int gg = g0 + wm * 32 + si * 16 + mbase + v;
                if (gg < G_DIM) {
                    const size_t gn = (size_t)gg * N_DIM + ng;
                    float x = (acc[v] + wb[gn] + db[gn]) * 0.5f;
                    out[((size_t)b * G_DIM + gg) * (size_t)N_DIM + ng] =
                        (x > 0.0f) ? x : LEAKY_ALPHA * x;
                }
            }
        }
    }
}

extern "C" void kernel_launch(void* const* d_in, const int* in_sizes, int n_in,
                              void* d_out, int out_size, void* d_ws, size_t ws_size,
                              hipStream_t stream) {
    (void)in_sizes; (void)n_in; (void)out_size; (void)d_ws; (void)ws_size;
    const float* inputs = (const float*)d_in[0];
    const int*   week   = (const int*)d_in[1];
    const int*   date   = (const int*)d_in[2];
    const float* week_w = (const float*)d_in[3];
    const float* week_b = (const float*)d_in[4];
    const float* day_w  = (const float*)d_in[5];
    const float* day_b  = (const float*)d_in[6];
    float* out = (float*)d_out;

    dim3 grid(N_DIM / BN, (G_DIM + BM - 1) / BM, B_DIM);  // (12, 3, 64)
    dim3 block(256);
    talinear_wmma_kernel<<<grid, block, 0, stream>>>(
        inputs, week, date, week_w, week_b, day_w, day_b, out);
}